// VectorQuantizer_62629213110906
// MI455X (gfx1250) — compile-verified
//
#include <hip/hip_runtime.h>

typedef __attribute__((ext_vector_type(2))) float v2f;
typedef __attribute__((ext_vector_type(8))) float v8f;

#define K_CB        1024
#define D_DIM       256
#define ROWS_PB     128          // vectors per block
#define NT          256          // threads per block (8 wave32)
#define CBS_STRIDE  260          // padded LDS stride (even, conflict-free for b64 reads)
#define XQ_ELEMS    8388608      // 32*256*32*32
#define LOSS_OFF    XQ_ELEMS
#define IDX_OFF     (XQ_ELEMS + 1)

__global__ __launch_bounds__(NT) void vq_main_kernel(
    const float* __restrict__ x,     // [32,256,32,32]
    const float* __restrict__ cb,    // [1024,256]
    float* __restrict__ out,         // xq(8388608) + loss(1) + idx(32768)
    float* __restrict__ ws)          // 256 partial loss sums
{
    __shared__ float cbs[16 * CBS_STRIDE];   // codebook tile, [j][d] padded
    __shared__ float esq_part[NT];
    __shared__ float esq_tile[16];
    __shared__ float redv[ROWS_PB * 16];
    __shared__ int   redi[ROWS_PB * 16];
    __shared__ int   kst[ROWS_PB];
    __shared__ float sred[NT];

    const int t    = threadIdx.x;
    const int lane = t & 31;
    const int w    = t >> 5;         // wave id 0..7
    const int half = lane >> 4;      // 0: lanes 0-15, 1: lanes 16-31
    const int jm   = lane & 15;

    const int n0 = blockIdx.x * ROWS_PB;     // first vector of this block
    const int b  = n0 >> 10;                 // batch image (128 | 1024)
    const int p0 = n0 & 1023;                // h*32+w offset inside image
    const float* xg = x + (size_t)b * (D_DIM * 1024) + p0;   // x_e[n0+r][d] = xg[d*1024 + r]
    const float* xw = xg + (w * 16 + jm);    // this lane's A row

    float minval[8];
    int   minidx[8];
    #pragma unroll
    for (int v = 0; v < 8; ++v) { minval[v] = 3.4e38f; minidx[v] = 0; }

    const int jload = t >> 4;          // codebook tile row this thread stages
    const int dbase = (t & 15) << 4;   // 16-float chunk it stages

    for (int kt = 0; kt < 64; ++kt) {
        const int kbase = kt << 4;
        __syncthreads();   // previous tile fully consumed

        // --- stage codebook tile [16 x 256] into LDS, accumulate ||e||^2 partials ---
        {
            const float* src = cb + (size_t)(kbase + jload) * D_DIM + dbase;
            float* dst = &cbs[jload * CBS_STRIDE + dbase];
            float ss = 0.0f;
            #pragma unroll
            for (int q = 0; q < 4; ++q) {
                float4 vq = *(const float4*)(src + q * 4);
                *(float4*)(dst + q * 4) = vq;
                ss += vq.x * vq.x + vq.y * vq.y + vq.z * vq.z + vq.w * vq.w;
            }
            esq_part[t] = ss;
        }
        __syncthreads();
        if (t < 16) {
            float acc = 0.0f;
            #pragma unroll
            for (int i = 0; i < 16; ++i) acc += esq_part[t * 16 + i];
            esq_tile[t] = acc;
        }
        __syncthreads();

        // --- 16x16 score tile: 64 x V_WMMA_F32_16X16X4_F32 over D=256 ---
        v8f c = {0.f, 0.f, 0.f, 0.f, 0.f, 0.f, 0.f, 0.f};
        #pragma unroll 8
        for (int s = 0; s < 64; ++s) {
            const int d = (s << 2) + (half << 1);
            v2f a, bv;
            a.x  = xw[(size_t)d * 1024];            // A: x row (coalesced across lanes)
            a.y  = xw[(size_t)(d + 1) * 1024];
            bv.x = cbs[jm * CBS_STRIDE + d];        // B: codebook column from LDS
            bv.y = cbs[jm * CBS_STRIDE + d + 1];
            c = __builtin_amdgcn_wmma_f32_16x16x4_f32(
                    false, a, false, bv, (short)0, c, false, false);
        }

        // --- running argmin of (||e||^2 - 2 e.x) ---
        const float esq = esq_tile[jm];
        const int   kk  = kbase + jm;
        #pragma unroll
        for (int v = 0; v < 8; ++v) {
            float dv = esq - 2.0f * c[v];
            if (dv < minval[v]) { minval[v] = dv; minidx[v] = kk; }
        }
    }

    // --- cross-lane argmin via LDS (C layout: row = v + 8*half, col = jm) ---
    __syncthreads();
    #pragma unroll
    for (int v = 0; v < 8; ++v) {
        int r = w * 16 + v + 8 * half;          // local row 0..127
        redv[r * 16 + jm] = minval[v];
        redi[r * 16 + jm] = minidx[v];
    }
    __syncthreads();
    if (t < ROWS_PB) {
        float bvv = redv[t * 16];
        int   bii = redi[t * 16];
        #pragma unroll
        for (int i = 1; i < 16; ++i) {
            float vv = redv[t * 16 + i];
            if (vv < bvv) { bvv = vv; bii = redi[t * 16 + i]; }
        }
        kst[t] = bii;
        out[IDX_OFF + n0 + t] = (float)bii;     // latent index as float
    }
    __syncthreads();

    // --- write x_q in [B,D,H,W] layout + accumulate squared error ---
    float acc = 0.0f;
    float* outq = out + (size_t)b * (D_DIM * 1024) + p0;
    const int r4 = (t & 31) << 2;               // 4 consecutive rows per thread
    for (int it = 0; it < 32; ++it) {
        const int d = (t >> 5) + (it << 3);
        float4 xv = *(const float4*)(xg + (size_t)d * 1024 + r4);
        float4 qv;
        qv.x = cb[(size_t)kst[r4 + 0] * D_DIM + d];
        qv.y = cb[(size_t)kst[r4 + 1] * D_DIM + d];
        qv.z = cb[(size_t)kst[r4 + 2] * D_DIM + d];
        qv.w = cb[(size_t)kst[r4 + 3] * D_DIM + d];
        *(float4*)(outq + (size_t)d * 1024 + r4) = qv;
        float dx = qv.x - xv.x, dy = qv.y - xv.y;
        float dz = qv.z - xv.z, dw = qv.w - xv.w;
        acc += dx * dx + dy * dy + dz * dz + dw * dw;
    }
    sred[t] = acc;
    __syncthreads();
    for (int off = NT >> 1; off > 0; off >>= 1) {   // deterministic tree
        if (t < off) sred[t] += sred[t + off];
        __syncthreads();
    }
    if (t == 0) ws[blockIdx.x] = sred[0];
}

__global__ __launch_bounds__(NT) void vq_loss_kernel(const float* __restrict__ ws,
                                                     float* __restrict__ out_loss)
{
    __shared__ float s[NT];
    const int t = threadIdx.x;
    s[t] = ws[t];
    __syncthreads();
    for (int off = NT >> 1; off > 0; off >>= 1) {
        if (t < off) s[t] += s[t + off];
        __syncthreads();
    }
    // loss = mean((xq-x)^2) + 0.5*mean((xq-x)^2) = 1.5 * sum / 8388608
    if (t == 0) *out_loss = 1.5f * s[0] / 8388608.0f;
}

extern "C" void kernel_launch(void* const* d_in, const int* in_sizes, int n_in,
                              void* d_out, int out_size, void* d_ws, size_t ws_size,
                              hipStream_t stream) {
    const float* x  = (const float*)d_in[0];   // [32,256,32,32] f32
    const float* cb = (const float*)d_in[1];   // [1024,256] f32
    float* out = (float*)d_out;
    float* ws  = (float*)d_ws;

    vq_main_kernel<<<dim3(256), dim3(NT), 0, stream>>>(x, cb, out, ws);
    vq_loss_kernel<<<dim3(1), dim3(NT), 0, stream>>>(ws, out + LOSS_OFF);
}